// AdditiveAttention_17102559773048
// MI455X (gfx1250) — compile-verified
//
#include <hip/hip_runtime.h>
#include <hip/hip_bf16.h>

#define B_  32
#define Q_  64
#define K_  512
#define H_  64
#define V_  64
#define NEG_ (-1e6f)
#define QG_ 4            // q rows per score block

typedef __attribute__((ext_vector_type(2))) float v2f;
typedef __attribute__((ext_vector_type(8))) float v8f;

__device__ __forceinline__ float fast_tanh(float x) {
#if __has_builtin(__builtin_amdgcn_tanhf)
    return __builtin_amdgcn_tanhf(x);   // gfx1250 v_tanh_f32
#else
    float e = __expf(2.0f * x);
    return 1.0f - 2.0f / (e + 1.0f);
#endif
}

// Store a 16x16 f32 C/D tile (8 VGPRs per lane) to row-major C with leading dim ldc.
// Lane l: N = l%16 ; VGPR r holds M = (l/16)*8 + r.
__device__ __forceinline__ void wmma_store_tile(v8f acc, float* __restrict__ C,
                                                int ldc, int lane) {
    const int n = lane & 15;
    const int mbase = (lane >> 4) << 3;
#pragma unroll
    for (int r = 0; r < 8; ++r)
        C[(size_t)(mbase + r) * ldc + n] = acc[r];
}

// ---------------------------------------------------------------------------
// Projection GEMM: Y[M,64] = X[M,64] @ W[64,64], one wave per 16x16 tile.
// ---------------------------------------------------------------------------
__global__ void __launch_bounds__(128)
proj_gemm_kernel(const float* __restrict__ X, const float* __restrict__ W,
                 float* __restrict__ Y) {
    const int lane = threadIdx.x & 31;
    const int wave = threadIdx.x >> 5;
    const int m0 = blockIdx.x * 16;
    const int n0 = wave * 16;
    const int rc   = lane & 15;            // A-row / B-col for this lane
    const int koff = (lane >> 4) << 1;     // 0 or 2

    const float* A  = X + (size_t)(m0 + rc) * H_;
    const float* Bp = W + n0 + rc;

    v8f acc = {};
#pragma unroll
    for (int kk = 0; kk < H_; kk += 4) {
        v2f a, b;
        a.x = A[kk + koff];
        a.y = A[kk + koff + 1];
        b.x = Bp[(size_t)(kk + koff) * H_];
        b.y = Bp[(size_t)(kk + koff + 1) * H_];
        acc = __builtin_amdgcn_wmma_f32_16x16x4_f32(
            false, a, false, b, (short)0, acc, false, false);
    }
    wmma_store_tile(acc, Y + (size_t)m0 * H_ + n0, H_, lane);
}

// ---------------------------------------------------------------------------
// Fused additive scores + masked softmax, q-grouped.
// One block per (b, group of QG_ q-rows); 256 threads; thread t owns keys
// t and t+256. Each loaded k-row chunk feeds QG_ q accumulators, cutting
// k_proj L2 traffic by QG_ x. tanh count is invariant (the true bound).
// ---------------------------------------------------------------------------
__global__ void __launch_bounds__(256)
scores_softmax_kernel(const float* __restrict__ qproj,
                      const float* __restrict__ kproj,
                      const float* __restrict__ wv,
                      const int*   __restrict__ valid_lens,
                      float*       __restrict__ attn) {
    __shared__ float qs[QG_][H_];
    __shared__ float wls[H_];
    __shared__ float red[QG_][8];
    __shared__ float bcast[2 * QG_];

    const int tid  = threadIdx.x;
    const int lane = tid & 31;
    const int wid  = tid >> 5;
    const int b    = blockIdx.x / (Q_ / QG_);
    const int q0   = (blockIdx.x % (Q_ / QG_)) * QG_;

    // 256 threads load QG_*H_ = 256 q-proj values; first 64 also load wv.
    qs[tid >> 6][tid & 63] = qproj[((size_t)b * Q_ + q0 + (tid >> 6)) * H_ + (tid & 63)];
    if (tid < H_) wls[tid] = wv[tid];
    __syncthreads();

    const int vlen = valid_lens[b];
    const int k0 = tid;
    const int k1 = tid + 256;
    const float* kp0 = kproj + ((size_t)b * K_ + k0) * H_;
    const float* kp1 = kp0 + (size_t)256 * H_;

    float s0[QG_] = {0.f, 0.f, 0.f, 0.f};
    float s1[QG_] = {0.f, 0.f, 0.f, 0.f};
#pragma unroll 4
    for (int h = 0; h < H_; h += 4) {
        float4 wq = *(const float4*)(wls + h);
        float4 ka = *(const float4*)(kp0 + h);
        float4 kb = *(const float4*)(kp1 + h);
#pragma unroll
        for (int j = 0; j < QG_; ++j) {
            float4 qv = *(const float4*)(&qs[j][h]);
            s0[j] += wq.x * fast_tanh(qv.x + ka.x);
            s0[j] += wq.y * fast_tanh(qv.y + ka.y);
            s0[j] += wq.z * fast_tanh(qv.z + ka.z);
            s0[j] += wq.w * fast_tanh(qv.w + ka.w);
            s1[j] += wq.x * fast_tanh(qv.x + kb.x);
            s1[j] += wq.y * fast_tanh(qv.y + kb.y);
            s1[j] += wq.z * fast_tanh(qv.z + kb.z);
            s1[j] += wq.w * fast_tanh(qv.w + kb.w);
        }
    }
#pragma unroll
    for (int j = 0; j < QG_; ++j) {
        if (k0 >= vlen) s0[j] = NEG_;
        if (k1 >= vlen) s1[j] = NEG_;
    }

    // per-row block max (4 rows reduced in parallel)
    float m[QG_];
#pragma unroll
    for (int j = 0; j < QG_; ++j) m[j] = fmaxf(s0[j], s1[j]);
#pragma unroll
    for (int off = 16; off; off >>= 1)
#pragma unroll
        for (int j = 0; j < QG_; ++j) m[j] = fmaxf(m[j], __shfl_xor(m[j], off));
    if (lane == 0)
#pragma unroll
        for (int j = 0; j < QG_; ++j) red[j][wid] = m[j];
    __syncthreads();
    if (tid < QG_) {
        float mm = red[tid][0];
#pragma unroll
        for (int i = 1; i < 8; ++i) mm = fmaxf(mm, red[tid][i]);
        bcast[tid] = mm;
    }
    __syncthreads();

    float e0[QG_], e1[QG_], sum[QG_];
#pragma unroll
    for (int j = 0; j < QG_; ++j) {
        const float mj = bcast[j];
        e0[j] = __expf(s0[j] - mj);
        e1[j] = __expf(s1[j] - mj);
        sum[j] = e0[j] + e1[j];
    }
#pragma unroll
    for (int off = 16; off; off >>= 1)
#pragma unroll
        for (int j = 0; j < QG_; ++j) sum[j] += __shfl_xor(sum[j], off);
    __syncthreads();                 // red[] reuse barrier
    if (lane == 0)
#pragma unroll
        for (int j = 0; j < QG_; ++j) red[j][wid] = sum[j];
    __syncthreads();
    if (tid < QG_) {
        float ss = 0.0f;
#pragma unroll
        for (int i = 0; i < 8; ++i) ss += red[tid][i];
        bcast[QG_ + tid] = 1.0f / ss;
    }
    __syncthreads();

#pragma unroll
    for (int j = 0; j < QG_; ++j) {
        const float inv = bcast[QG_ + j];
        float* arow = attn + ((size_t)b * Q_ + q0 + j) * K_;
        arow[k0] = e0[j] * inv;
        arow[k1] = e1[j] * inv;
    }
}

// ---------------------------------------------------------------------------
// out[b] = attn[b] (64x512) @ values[b] (512x64); one wave per 16x16 tile.
// ---------------------------------------------------------------------------
__global__ void __launch_bounds__(128)
av_gemm_kernel(const float* __restrict__ attn, const float* __restrict__ values,
               float* __restrict__ out) {
    const int lane = threadIdx.x & 31;
    const int wave = threadIdx.x >> 5;
    const int b  = blockIdx.x >> 2;
    const int mt = blockIdx.x & 3;
    const int m0 = mt * 16;
    const int n0 = wave * 16;
    const int rc   = lane & 15;
    const int koff = (lane >> 4) << 1;

    const float* A  = attn + ((size_t)b * Q_ + m0 + rc) * K_;
    const float* Bp = values + (size_t)b * K_ * V_ + n0 + rc;

    v8f acc = {};
#pragma unroll 4
    for (int kk = 0; kk < K_; kk += 4) {
        v2f a, bv;
        a.x  = A[kk + koff];
        a.y  = A[kk + koff + 1];
        bv.x = Bp[(size_t)(kk + koff) * V_];
        bv.y = Bp[(size_t)(kk + koff + 1) * V_];
        acc = __builtin_amdgcn_wmma_f32_16x16x4_f32(
            false, a, false, bv, (short)0, acc, false, false);
    }
    wmma_store_tile(acc, out + ((size_t)b * Q_ + m0) * V_ + n0, V_, lane);
}

extern "C" void kernel_launch(void* const* d_in, const int* in_sizes, int n_in,
                              void* d_out, int out_size, void* d_ws, size_t ws_size,
                              hipStream_t stream) {
    const float* queries = (const float*)d_in[0];   // (B,Q,H)
    const float* keys    = (const float*)d_in[1];   // (B,K,H)
    const float* values  = (const float*)d_in[2];   // (B,K,V)
    const int*   vlens   = (const int*)  d_in[3];   // (B,)
    const float* Wq      = (const float*)d_in[4];   // (H,H)
    const float* Wk      = (const float*)d_in[5];   // (H,H)
    const float* wv      = (const float*)d_in[6];   // (H,)
    float* out = (float*)d_out;                     // (B,Q,V)

    float* qproj = (float*)d_ws;                          // B*Q*H   = 131072 f32
    float* kproj = qproj + (size_t)B_ * Q_ * H_;          // B*K*H   = 1048576 f32
    float* attn  = kproj + (size_t)B_ * K_ * H_;          // B*Q*K   = 1048576 f32

    // 1) projections (WMMA f32)
    proj_gemm_kernel<<<(B_ * Q_) / 16, 128, 0, stream>>>(queries, Wq, qproj);
    proj_gemm_kernel<<<(B_ * K_) / 16, 128, 0, stream>>>(keys,    Wk, kproj);
    // 2) additive scores + tanh + masked softmax (VALU/tanh bound, q-grouped)
    scores_softmax_kernel<<<B_ * (Q_ / QG_), 256, 0, stream>>>(qproj, kproj, wv, vlens, attn);
    // 3) attention-weighted values (WMMA f32)
    av_gemm_kernel<<<B_ * 4, 128, 0, stream>>>(attn, values, out);
}